// GAU_71038759075983
// MI455X (gfx1250) — compile-verified
//
#include <hip/hip_runtime.h>

// ---------------------------------------------------------------------------
// GAU block for MI455X (gfx1250, wave32, WMMA).
// Math simplification: softmax rows sum to exactly 1 => sum(attn,-1)==1, so
// the whole qk/attention branch is the identity on v (W_qk branch is dead).
// Pipeline: LN -> [GEMM 1024->4096 + bias + silu + v*gate fused, f16 WMMA]
//           -> [GEMM 2048->1024 + bias + residual, f16 WMMA] -> fp32 out.
// WMMA via inline asm with tied "+v" accumulator (D==C in one tuple).
// __launch_bounds__(256, 1) relaxes the waves-per-EU-derived VGPR budget:
// the default budget made regalloc split accumulator live ranges at the
// phase boundaries (v_mov_b64 storms in rounds 2-6).  Weights pre-packed in
// 32-K column-major slabs; A fragments direct from global; single LDS buffer,
// branch-free two-barrier loop; load/compute phases split by sched_barrier.
// ---------------------------------------------------------------------------

typedef __attribute__((ext_vector_type(16))) _Float16 v16h;
typedef __attribute__((ext_vector_type(8)))  _Float16 v8h;
typedef __attribute__((ext_vector_type(4)))  _Float16 v4h;
typedef __attribute__((ext_vector_type(8)))  float    v8f;

struct F16Frag { v8h lo, hi; };
static_assert(sizeof(F16Frag) == sizeof(v16h), "frag size");

#define ROWS   16384      // B*N = 4*4096
#define DIM    1024
#define HIDDEN 2048
#define LN_EPS 1e-5f
#define LDSTR  40         // 32 halfs + 8 pad; 80B rows keep b128 16B-aligned

__device__ __forceinline__ float silu_f(float v) {
  return v / (1.0f + __expf(-v));
}

__device__ __forceinline__ v16h make_frag(const _Float16* p) {
  F16Frag f;
  f.lo = *reinterpret_cast<const v8h*>(p);
  f.hi = *reinterpret_cast<const v8h*>(p + 16);
  return __builtin_bit_cast(v16h, f);
}

// Tied-accumulator WMMA: forces D == C so the accumulator lives in one
// register tuple (non-volatile: result is used, so it is not eliminated).
__device__ __forceinline__ void wmma_acc(v8f& c, const v16h& a, const v16h& b) {
  asm("v_wmma_f32_16x16x32_f16 %0, %1, %2, %0"
      : "+v"(c)
      : "v"(a), "v"(b));
}

// ---------------------------------------------------------------------------
// Kernel 1: LayerNorm over DIM, write f16 activations (WMMA A operand).
// ---------------------------------------------------------------------------
__global__ __launch_bounds__(256) void ln_cast_kernel(
    const float* __restrict__ x, const float* __restrict__ g,
    const float* __restrict__ b, _Float16* __restrict__ out)
{
  __shared__ float rs[256];
  __shared__ float rq[256];
  const int row = blockIdx.x;
  const int t   = threadIdx.x;
  const float4 v = reinterpret_cast<const float4*>(x + (size_t)row * DIM)[t];
  rs[t] = v.x + v.y + v.z + v.w;
  rq[t] = v.x * v.x + v.y * v.y + v.z * v.z + v.w * v.w;
  __syncthreads();
  for (int off = 128; off > 0; off >>= 1) {
    if (t < off) { rs[t] += rs[t + off]; rq[t] += rq[t + off]; }
    __syncthreads();
  }
  const float mean = rs[0] * (1.0f / DIM);
  const float var  = rq[0] * (1.0f / DIM) - mean * mean;
  const float inv  = rsqrtf(var + LN_EPS);
  const float4 gg = reinterpret_cast<const float4*>(g)[t];
  const float4 bb = reinterpret_cast<const float4*>(b)[t];
  v4h o;
  o[0] = (_Float16)((v.x - mean) * inv * gg.x + bb.x);
  o[1] = (_Float16)((v.y - mean) * inv * gg.y + bb.y);
  o[2] = (_Float16)((v.z - mean) * inv * gg.z + bb.z);
  o[3] = (_Float16)((v.w - mean) * inv * gg.w + bb.w);
  *reinterpret_cast<v4h*>(&out[(size_t)row * DIM + t * 4]) = o;
}

// ---------------------------------------------------------------------------
// Kernel 2: weight f32 -> f16 + slab packing.
// Wp[((k/32)*N + n)*32 + (k&31)] = (f16) W[k*N + n]
// ---------------------------------------------------------------------------
__global__ __launch_bounds__(256) void cvt_pack_w_kernel(
    const float* __restrict__ in, _Float16* __restrict__ out, int K, int N)
{
  const int i = blockIdx.x * 256 + threadIdx.x;
  if (i >= K * N) return;
  const int k = i / N;
  const int n = i - k * N;
  out[(((size_t)(k >> 5)) * N + n) * 32 + (k & 31)] = (_Float16)in[i];
}

// ---------------------------------------------------------------------------
// Kernel 3: fused  h = silu(A @ W_hidden + b);  VG = h[:, :H] * h[:, H:]
// WG: 128 rows x 64 vg-cols (stages 128 h-cols: 64 v + 64 gate).
// Wave (waveM, waveN): rows [32*waveM, +32), vg cols [n0+32*waveN, +32)
// -> nt 0,1 = v tiles; nt 2,3 = gate tiles; 8 WMMAs / K-step / wave.
// ---------------------------------------------------------------------------
__global__ __launch_bounds__(256, 1) void gemm_hidden_kernel(
    const _Float16* __restrict__ A,     // ROWS x DIM        (f16 row-major)
    const _Float16* __restrict__ Wp,    // packed DIM x 4096 (f16 slabs)
    const float*    __restrict__ bias,  // 2*HIDDEN
    _Float16*       __restrict__ VG)    // ROWS x HIDDEN
{
  constexpr int LDB = 2 * HIDDEN;       // 4096
  constexpr int NSTEP = DIM / 32;       // 32
  __shared__ alignas(16) _Float16 lB[128 * LDSTR];

  const int t      = threadIdx.x;
  const int wave   = t >> 5;
  const int waveM  = wave & 3;          // 4 M positions x 32 rows
  const int waveN  = wave >> 2;         // 2 N positions x 32 vg cols
  const int lane   = t & 31;
  const int laneN  = lane & 15;
  const int laneHi = lane >> 4;
  const int m0     = blockIdx.y * 128;
  const int n0     = blockIdx.x * 64;   // vg column base in [0, HIDDEN)

  // B staging task: one staged column (0..63 = v, 64..127 = gate), one chunk.
  const int nl = t >> 1;                // 0..127 staged column
  const int ch = (t & 1) * 16;          // 0 or 16 (K offset)
  const int gcol = (nl < 64) ? (n0 + nl) : (HIDDEN + n0 + (nl - 64));
  const _Float16* wbase = &Wp[(size_t)gcol * 32 + ch];
  _Float16* sdst = &lB[0] + nl * LDSTR + ch;
  constexpr size_t slabStride = (size_t)LDB * 32;

  // A fragment base (rows are K-contiguous in global).
  const _Float16* abase =
      &A[(size_t)(m0 + waveM * 32 + laneN) * DIM + laneHi * 8];

  const v8f vzero = {0.f, 0.f, 0.f, 0.f, 0.f, 0.f, 0.f, 0.f};
  v8f a00 = vzero, a01 = vzero, a02 = vzero, a03 = vzero;
  v8f a10 = vzero, a11 = vzero, a12 = vzero, a13 = vzero;

  // Prologue: fetch slab 0 into registers.
  v8h w0 = *reinterpret_cast<const v8h*>(wbase);
  v8h w1 = *reinterpret_cast<const v8h*>(wbase + 8);

  const _Float16* lbc = &lB[0] + laneN * LDSTR + laneHi * 8;
  const int sc0 = waveN * 32;           // v tile 0
  const int sc1 = waveN * 32 + 16;      // v tile 1
  const int sc2 = 64 + waveN * 32;      // gate tile 0
  const int sc3 = 64 + waveN * 32 + 16; // gate tile 1

  for (int i = 0; i < NSTEP; ++i) {
    __syncthreads();                    // readers of lB (iter i-1) are done
    *reinterpret_cast<v8h*>(sdst)     = w0;
    *reinterpret_cast<v8h*>(sdst + 8) = w1;
    // Clamped next-slab fetch: branch-free; overlaps the compute phase.
    const int nx = (i + 1 < NSTEP) ? (i + 1) : i;
    const _Float16* wn = wbase + (size_t)nx * slabStride;
    w0 = *reinterpret_cast<const v8h*>(wn);
    w1 = *reinterpret_cast<const v8h*>(wn + 8);
    __syncthreads();                    // staged slab visible

    // ---- load phase: all fragments for this K-step ----
    const int k0 = i * 32;
    const v16h af0 = make_frag(abase + k0);
    const v16h af1 = make_frag(abase + (size_t)16 * DIM + k0);
    const v16h bf0 = make_frag(lbc + sc0 * LDSTR);
    const v16h bf1 = make_frag(lbc + sc1 * LDSTR);
    const v16h bf2 = make_frag(lbc + sc2 * LDSTR);
    const v16h bf3 = make_frag(lbc + sc3 * LDSTR);
    __builtin_amdgcn_sched_barrier(0);

    // ---- compute phase: 8 back-to-back WMMAs, tied accumulators ----
    wmma_acc(a00, af0, bf0);
    wmma_acc(a10, af1, bf0);
    wmma_acc(a01, af0, bf1);
    wmma_acc(a11, af1, bf1);
    wmma_acc(a02, af0, bf2);
    wmma_acc(a12, af1, bf2);
    wmma_acc(a03, af0, bf3);
    wmma_acc(a13, af1, bf3);
    __builtin_amdgcn_sched_barrier(0);
  }

  // Epilogue: bias + silu on v and gate, multiply, store f16 vg.
  const v8f accv[2][2] = { { a00, a01 }, { a10, a11 } };
  const v8f accg[2][2] = { { a02, a03 }, { a12, a13 } };
#pragma unroll
  for (int nt = 0; nt < 2; ++nt) {
    const int   gn = n0 + waveN * 32 + nt * 16 + laneN;
    const float bv = bias[gn];
    const float bg = bias[gn + HIDDEN];
#pragma unroll
    for (int mi = 0; mi < 2; ++mi) {
#pragma unroll
      for (int r = 0; r < 8; ++r) {
        const int row = m0 + waveM * 32 + mi * 16 + laneHi * 8 + r;
        const float hv = silu_f(accv[mi][nt][r] + bv);
        const float hg = silu_f(accg[mi][nt][r] + bg);
        VG[(size_t)row * HIDDEN + gn] = (_Float16)(hv * hg);
      }
    }
  }
}

// ---------------------------------------------------------------------------
// Kernel 4: out = VG @ W_out + b_out + x   (fp32 output, residual fused)
// WG: 128 rows x 128 cols; wave (waveM, waveN): 32 rows x 64 cols.
// ---------------------------------------------------------------------------
__global__ __launch_bounds__(256, 1) void gemm_out_kernel(
    const _Float16* __restrict__ VG,    // ROWS x HIDDEN (f16 row-major)
    const _Float16* __restrict__ Wp,    // packed HIDDEN x DIM (f16 slabs)
    const float*    __restrict__ bias,  // DIM
    const float*    __restrict__ xres,  // ROWS x DIM
    float*          __restrict__ out)   // ROWS x DIM
{
  constexpr int NSTEP = HIDDEN / 32;    // 64
  __shared__ alignas(16) _Float16 lB[128 * LDSTR];

  const int t      = threadIdx.x;
  const int wave   = t >> 5;
  const int waveM  = wave & 3;
  const int waveN  = wave >> 2;
  const int lane   = t & 31;
  const int laneN  = lane & 15;
  const int laneHi = lane >> 4;
  const int m0     = blockIdx.y * 128;
  const int n0     = blockIdx.x * 128;

  const int nl = t >> 1;
  const int ch = (t & 1) * 16;
  const _Float16* wbase = &Wp[(size_t)(n0 + nl) * 32 + ch];
  _Float16* sdst = &lB[0] + nl * LDSTR + ch;
  constexpr size_t slabStride = (size_t)DIM * 32;

  const _Float16* abase =
      &VG[(size_t)(m0 + waveM * 32 + laneN) * HIDDEN + laneHi * 8];

  const v8f vzero = {0.f, 0.f, 0.f, 0.f, 0.f, 0.f, 0.f, 0.f};
  v8f a00 = vzero, a01 = vzero, a02 = vzero, a03 = vzero;
  v8f a10 = vzero, a11 = vzero, a12 = vzero, a13 = vzero;

  v8h w0 = *reinterpret_cast<const v8h*>(wbase);
  v8h w1 = *reinterpret_cast<const v8h*>(wbase + 8);

  const _Float16* lbc = &lB[0] + laneN * LDSTR + laneHi * 8;
  const int sc0 = waveN * 64;
  const int sc1 = waveN * 64 + 16;
  const int sc2 = waveN * 64 + 32;
  const int sc3 = waveN * 64 + 48;

  for (int i = 0; i < NSTEP; ++i) {
    __syncthreads();
    *reinterpret_cast<v8h*>(sdst)     = w0;
    *reinterpret_cast<v8h*>(sdst + 8) = w1;
    const int nx = (i + 1 < NSTEP) ? (i + 1) : i;
    const _Float16* wn = wbase + (size_t)nx * slabStride;
    w0 = *reinterpret_cast<const v8h*>(wn);
    w1 = *reinterpret_cast<const v8h*>(wn + 8);
    __syncthreads();

    const int k0 = i * 32;
    const v16h af0 = make_frag(abase + k0);
    const v16h af1 = make_frag(abase + (size_t)16 * HIDDEN + k0);
    const v16h bf0 = make_frag(lbc + sc0 * LDSTR);
    const v16h bf1 = make_frag(lbc + sc1 * LDSTR);
    const v16h bf2 = make_frag(lbc + sc2 * LDSTR);
    const v16h bf3 = make_frag(lbc + sc3 * LDSTR);
    __builtin_amdgcn_sched_barrier(0);

    wmma_acc(a00, af0, bf0);
    wmma_acc(a10, af1, bf0);
    wmma_acc(a01, af0, bf1);
    wmma_acc(a11, af1, bf1);
    wmma_acc(a02, af0, bf2);
    wmma_acc(a12, af1, bf2);
    wmma_acc(a03, af0, bf3);
    wmma_acc(a13, af1, bf3);
    __builtin_amdgcn_sched_barrier(0);
  }

  const v8f acc[2][4] = { { a00, a01, a02, a03 }, { a10, a11, a12, a13 } };
#pragma unroll
  for (int nt = 0; nt < 4; ++nt) {
    const int   gn = n0 + waveN * 64 + nt * 16 + laneN;
    const float bv = bias[gn];
#pragma unroll
    for (int mi = 0; mi < 2; ++mi) {
#pragma unroll
      for (int r = 0; r < 8; ++r) {
        const int row = m0 + waveM * 32 + mi * 16 + laneHi * 8 + r;
        const size_t o = (size_t)row * DIM + gn;
        out[o] = acc[mi][nt][r] + bv + xres[o];
      }
    }
  }
}

// ---------------------------------------------------------------------------
// Host launcher.  Workspace (~108 MB):
//   [0,32MB)    A16  : f16 normed activations  (16384 x 1024)
//   [32,40MB)   Wh16 : f16 packed W_hidden     (1024 x 4096)
//   [40,44MB)   Wo16 : f16 packed W_out        (2048 x 1024)
//   [44,108MB)  VG   : f16 v*gate              (16384 x 2048)
// ---------------------------------------------------------------------------
extern "C" void kernel_launch(void* const* d_in, const int* in_sizes, int n_in,
                              void* d_out, int out_size, void* d_ws, size_t ws_size,
                              hipStream_t stream)
{
  (void)in_sizes; (void)n_in; (void)out_size; (void)ws_size;

  const float* x    = (const float*)d_in[0];
  const float* ln_g = (const float*)d_in[1];
  const float* ln_b = (const float*)d_in[2];
  const float* Wh   = (const float*)d_in[3];
  const float* bh   = (const float*)d_in[4];
  // d_in[5..8] (W_qk, b_qk, os_gamma, os_beta) are mathematically dead:
  // sum(softmax(.), -1) == 1, so the attention branch is the identity on v.
  const float* Wo   = (const float*)d_in[9];
  const float* bo   = (const float*)d_in[10];
  float* out = (float*)d_out;

  char* ws = (char*)d_ws;
  const size_t szA  = (size_t)ROWS * DIM * sizeof(_Float16);
  const size_t szWh = (size_t)DIM * 2 * HIDDEN * sizeof(_Float16);
  const size_t szWo = (size_t)HIDDEN * DIM * sizeof(_Float16);
  _Float16* A16  = (_Float16*)(ws);
  _Float16* Wh16 = (_Float16*)(ws + szA);
  _Float16* Wo16 = (_Float16*)(ws + szA + szWh);
  _Float16* VG   = (_Float16*)(ws + szA + szWh + szWo);

  ln_cast_kernel<<<ROWS, 256, 0, stream>>>(x, ln_g, ln_b, A16);

  const int nWh = DIM * 2 * HIDDEN;
  const int nWo = HIDDEN * DIM;
  cvt_pack_w_kernel<<<(nWh + 255) / 256, 256, 0, stream>>>(Wh, Wh16, DIM, 2 * HIDDEN);
  cvt_pack_w_kernel<<<(nWo + 255) / 256, 256, 0, stream>>>(Wo, Wo16, HIDDEN, DIM);

  gemm_hidden_kernel<<<dim3(HIDDEN / 64, ROWS / 128), 256, 0, stream>>>(
      A16, Wh16, bh, VG);

  gemm_out_kernel<<<dim3(DIM / 128, ROWS / 128), 256, 0, stream>>>(
      VG, Wo16, bo, x, out);
}